// layer_N_c_d_d_to_N_d_d_3_LogEig_softmax2_81252191306557
// MI455X (gfx1250) — compile-verified
//
#include <hip/hip_runtime.h>
#include <hip/hip_bf16.h>
#include <math.h>

typedef float v2f __attribute__((ext_vector_type(2)));
typedef float v8f __attribute__((ext_vector_type(8)));

#define WMMA_F32(a, b, c) \
    __builtin_amdgcn_wmma_f32_16x16x4_f32(false, (a), false, (b), (short)0, (c), false, false)

// ---------------------------------------------------------------------------
// Kernel 1: scalar embedding MLP (tiny). Writes em3[10] to workspace.
// ---------------------------------------------------------------------------
__global__ __launch_bounds__(128) void emb_kernel(
    const int* __restrict__ Mp,
    const float* __restrict__ we1, const float* __restrict__ be1,
    const float* __restrict__ we2, const float* __restrict__ be2,
    const float* __restrict__ we3, const float* __restrict__ be3,
    const float* __restrict__ lng, const float* __restrict__ lnb,
    float* __restrict__ em3out)
{
    __shared__ float em[10], eln[10], em2[100];
    const int tid = threadIdx.x;
    const float md0 = (float)(*Mp) / 500.0f;
    const float md1 = 64.0f / 100.0f;
    if (tid < 10)
        em[tid] = md0 * we1[tid] + md1 * we1[10 + tid] + be1[tid];
    __syncthreads();
    if (tid < 10) {
        float mu = 0.f;
        for (int j = 0; j < 10; ++j) mu += em[j];
        mu *= 0.1f;
        float var = 0.f;
        for (int j = 0; j < 10; ++j) { float d = em[j] - mu; var += d * d; }
        var *= 0.1f;
        eln[tid] = (em[tid] - mu) / sqrtf(var + 1e-3f) * lng[tid] + lnb[tid];
    }
    __syncthreads();
    if (tid < 100) {
        float acc = be2[tid];
        for (int j = 0; j < 10; ++j) acc += eln[j] * we2[j * 100 + tid];
        em2[tid] = fmaxf(acc, 0.f);
    }
    __syncthreads();
    if (tid < 10) {
        float acc = em[tid] + be3[tid];
        for (int k = 0; k < 100; ++k) acc += em2[k] * we3[k * 10 + tid];
        em3out[tid] = acc;
    }
}

// ---------------------------------------------------------------------------
// Kernel 2: symmetric eig (parallel cyclic Jacobi) + V*diag(log l)*V^T via WMMA.
// One block (256 thr, 8 waves) per (n, channel) 64x64 matrix.
// Writes channel-last eu[n][i][k][ch] for ch in [0,22).
// ---------------------------------------------------------------------------
#define LD 68           // padded LDS row stride (floats) to avoid bank conflicts
#define NSWEEP 9

__global__ __launch_bounds__(256) void eig_kernel(
    const float* __restrict__ x, float* __restrict__ eu)
{
    __shared__ float As[64 * LD];
    __shared__ float Vs[64 * LD];
    __shared__ float logv_s[64];
    __shared__ float cs_s[32], sn_s[32];
    __shared__ int pp_s[32], qq_s[32];

    const int tid = threadIdx.x;
    const int mat = blockIdx.x;                 // 0 .. 5632-1
    const int n = mat / 22;
    const int ch = mat - n * 22;
    const size_t base = (size_t)mat * 4096;

    // Load + symmetrize; V = I
    #pragma unroll
    for (int k = 0; k < 16; ++k) {
        int idx = tid + k * 256;
        int i = idx >> 6, j = idx & 63;
        As[i * LD + j] = 0.5f * (x[base + idx] + x[base + (j << 6) + i]);
        Vs[i * LD + j] = (i == j) ? 1.0f : 0.0f;
    }
    __syncthreads();

    const int pair = tid >> 3;          // 0..31
    const int t8 = tid & 7;             // 0..7  -> 8 columns each

    for (int sweep = 0; sweep < NSWEEP; ++sweep) {
        for (int r = 0; r < 63; ++r) {
            // 1) compute disjoint rotation set (round-robin tournament)
            if (tid < 32) {
                int p, q;
                if (tid == 0) { p = 63; q = r; }
                else {
                    p = (r + tid) % 63;
                    q = (r - tid + 63) % 63;
                }
                if (p > q) { int t = p; p = q; q = t; }
                float apq = As[p * LD + q];
                float c = 1.f, s = 0.f;
                if (fabsf(apq) > 1e-12f) {
                    float app = As[p * LD + p];
                    float aqq = As[q * LD + q];
                    float tau = (aqq - app) / (2.0f * apq);
                    float tt = copysignf(1.0f, tau) /
                               (fabsf(tau) + sqrtf(1.0f + tau * tau));
                    c = 1.0f / sqrtf(1.0f + tt * tt);
                    s = tt * c;
                }
                cs_s[tid] = c; sn_s[tid] = s; pp_s[tid] = p; qq_s[tid] = q;
            }
            __syncthreads();

            const float c = cs_s[pair], s = sn_s[pair];
            const int p = pp_s[pair], q = qq_s[pair];

            // 2) row update: A <- J^T A
            #pragma unroll
            for (int jj = 0; jj < 8; ++jj) {
                int j = t8 * 8 + jj;
                float ap = As[p * LD + j], aq = As[q * LD + j];
                As[p * LD + j] = c * ap - s * aq;
                As[q * LD + j] = s * ap + c * aq;
            }
            __syncthreads();

            // 3) column update: A <- A J ; V <- V J
            #pragma unroll
            for (int jj = 0; jj < 8; ++jj) {
                int j = t8 * 8 + jj;
                float ap = As[j * LD + p], aq = As[j * LD + q];
                As[j * LD + p] = c * ap - s * aq;
                As[j * LD + q] = s * ap + c * aq;
                float vp = Vs[j * LD + p], vq = Vs[j * LD + q];
                Vs[j * LD + p] = c * vp - s * vq;
                Vs[j * LD + q] = s * vp + c * vq;
            }
            __syncthreads();
        }
    }

    // log-eigenvalues (clamped) from diagonal
    if (tid < 64)
        logv_s[tid] = logf(fmaxf(As[tid * LD + tid], 1e-4f));
    __syncthreads();

    // W = V * diag(log l)  (overwrite A in LDS)
    #pragma unroll
    for (int k = 0; k < 16; ++k) {
        int idx = tid + k * 256;
        int i = idx >> 6, j = idx & 63;
        As[i * LD + j] = Vs[i * LD + j] * logv_s[j];
    }
    __syncthreads();

    // L = W @ V^T via v_wmma_f32_16x16x4_f32. 16 tiles of 16x16, 2 per wave.
    const int lane = tid & 31, wave = tid >> 5;
    const int hi = lane >> 4, li = lane & 15;
    const int ksel = hi * 2;
    for (int t = 0; t < 2; ++t) {
        const int tIdx = wave + t * 8;
        const int rowBase = (tIdx >> 2) * 16;
        const int colBase = (tIdx & 3) * 16;
        v8f acc = {};
        #pragma unroll
        for (int kb = 0; kb < 16; ++kb) {
            const int k0 = kb * 4 + ksel;
            v2f a, b;
            a.x = As[(rowBase + li) * LD + k0];
            a.y = As[(rowBase + li) * LD + k0 + 1];
            b.x = Vs[(colBase + li) * LD + k0];     // B[k][col] = V[col][k]
            b.y = Vs[(colBase + li) * LD + k0 + 1];
            acc = WMMA_F32(a, b, acc);
        }
        #pragma unroll
        for (int v = 0; v < 8; ++v) {
            const int row = rowBase + v + hi * 8;
            const int col = colBase + li;
            eu[((size_t)((n * 64 + row) * 64 + col)) * 32 + ch] = acc[v];
        }
    }
}

// ---------------------------------------------------------------------------
// Kernel 3: broadcast em3 into channels 22..31 of eu.
// ---------------------------------------------------------------------------
__global__ __launch_bounds__(256) void fillemb_kernel(
    float* __restrict__ A, const float* __restrict__ em3)
{
    const size_t gid = (size_t)blockIdx.x * 256 + threadIdx.x;  // n*4096+p
    float* dst = A + gid * 32 + 22;
    #pragma unroll
    for (int e = 0; e < 10; ++e) dst[e] = em3[e];
}

// ---------------------------------------------------------------------------
// Kernel 4: per-(n,ch) mean / inv-std over 4096 spatial positions.
// ---------------------------------------------------------------------------
__global__ __launch_bounds__(256) void stats_kernel(
    const float* __restrict__ A, float* __restrict__ meanb,
    float* __restrict__ invb)
{
    __shared__ float s1[256], s2[256];
    const int tid = threadIdx.x;
    const int n = blockIdx.x >> 5, ch = blockIdx.x & 31;
    const float* base = A + (size_t)n * 4096 * 32 + ch;
    float sum = 0.f, sq = 0.f;
    for (int i = tid; i < 4096; i += 256) {
        float v = base[(size_t)i * 32];
        sum += v; sq += v * v;
    }
    s1[tid] = sum; s2[tid] = sq;
    __syncthreads();
    for (int st = 128; st > 0; st >>= 1) {
        if (tid < st) { s1[tid] += s1[tid + st]; s2[tid] += s2[tid + st]; }
        __syncthreads();
    }
    if (tid == 0) {
        float m = s1[0] * (1.0f / 4096.0f);
        float var = s2[0] * (1.0f / 4096.0f) - m * m;
        float sd = sqrtf(fmaxf(var, 0.f));
        meanb[blockIdx.x] = m;
        invb[blockIdx.x] = 1.0f / fmaxf(sd, 1e-3f);
    }
}

// ---------------------------------------------------------------------------
// Kernel 5: fused residual block, in place:
//   A <- A + relu(norm(A) @ wA + bA) @ wB + bB
// 128 threads (4 waves); wave handles 16 rows x 32 channels via WMMA f32.
// ---------------------------------------------------------------------------
__global__ __launch_bounds__(128) void resblock_kernel(
    float* __restrict__ A,
    const float* __restrict__ meanb, const float* __restrict__ invb,
    const float* __restrict__ wA, const float* __restrict__ bA,
    const float* __restrict__ wB, const float* __restrict__ bB)
{
    __shared__ float wAs[1024], wBs[1024];
    __shared__ float bAs[32], bBs[32], mean_s[32], inv_s[32];
    __shared__ float tS[4 * 16 * 33];   // per-wave relayout buffer (padded)

    const int tid = threadIdx.x;
    const int lane = tid & 31, wave = tid >> 5;
    const int hi = lane >> 4, li = lane & 15;
    const int ksel = hi * 2;
    const size_t blockRow = (size_t)blockIdx.x * 64;      // n*4096 + p
    const int n = (int)(blockRow >> 12);

    for (int i = tid; i < 1024; i += 128) { wAs[i] = wA[i]; wBs[i] = wB[i]; }
    if (tid < 32) {
        bAs[tid] = bA[tid]; bBs[tid] = bB[tid];
        mean_s[tid] = meanb[n * 32 + tid];
        inv_s[tid] = invb[n * 32 + tid];
    }
    __syncthreads();

    const size_t rowBase = blockRow + (size_t)wave * 16;
    const float* Arow = A + (rowBase + li) * 32;          // this lane's input row

    // t = relu(norm(A) @ wA + bA)
    v8f c0 = {}, c1 = {};
    #pragma unroll
    for (int kb = 0; kb < 8; ++kb) {
        const int k0 = kb * 4 + ksel;
        v2f a, b0, b1;
        a.x = (Arow[k0] - mean_s[k0]) * inv_s[k0];
        a.y = (Arow[k0 + 1] - mean_s[k0 + 1]) * inv_s[k0 + 1];
        b0.x = wAs[k0 * 32 + li];       b0.y = wAs[(k0 + 1) * 32 + li];
        b1.x = wAs[k0 * 32 + 16 + li];  b1.y = wAs[(k0 + 1) * 32 + 16 + li];
        c0 = WMMA_F32(a, b0, c0);
        c1 = WMMA_F32(a, b1, c1);
    }
    float* tw = tS + wave * (16 * 33);
    #pragma unroll
    for (int v = 0; v < 8; ++v) {
        const int row = v + hi * 8;
        tw[row * 33 + li]      = fmaxf(c0[v] + bAs[li], 0.f);
        tw[row * 33 + 16 + li] = fmaxf(c1[v] + bAs[16 + li], 0.f);
    }
    __syncthreads();

    // h = A + t @ wB + bB  (in place)
    v8f d0 = {}, d1 = {};
    #pragma unroll
    for (int kb = 0; kb < 8; ++kb) {
        const int k0 = kb * 4 + ksel;
        v2f a, b0, b1;
        a.x = tw[li * 33 + k0];
        a.y = tw[li * 33 + k0 + 1];
        b0.x = wBs[k0 * 32 + li];       b0.y = wBs[(k0 + 1) * 32 + li];
        b1.x = wBs[k0 * 32 + 16 + li];  b1.y = wBs[(k0 + 1) * 32 + 16 + li];
        d0 = WMMA_F32(a, b0, d0);
        d1 = WMMA_F32(a, b1, d1);
    }
    #pragma unroll
    for (int v = 0; v < 8; ++v) {
        const int row = v + hi * 8;
        const size_t g = (rowBase + row) * 32;
        A[g + li]      = A[g + li]      + d0[v] + bBs[li];
        A[g + 16 + li] = A[g + 16 + li] + d1[v] + bBs[16 + li];
    }
}

// ---------------------------------------------------------------------------
// Kernel 6: softmax(cov3 @ w). One thread per sample (N=256).
// ---------------------------------------------------------------------------
__global__ __launch_bounds__(256) void final_kernel(
    const float* __restrict__ cov, const float* __restrict__ w,
    float* __restrict__ out)
{
    const int nidx = threadIdx.x;
    const float* cv = cov + nidx * 32;
    float logit[7];
    #pragma unroll
    for (int k = 0; k < 7; ++k) logit[k] = 0.f;
    for (int chn = 0; chn < 32; ++chn) {
        const float v = cv[chn];
        #pragma unroll
        for (int k = 0; k < 7; ++k) logit[k] += v * w[chn * 7 + k];
    }
    float mx = logit[0];
    #pragma unroll
    for (int k = 1; k < 7; ++k) mx = fmaxf(mx, logit[k]);
    float se = 0.f;
    #pragma unroll
    for (int k = 0; k < 7; ++k) { logit[k] = expf(logit[k] - mx); se += logit[k]; }
    const float rs = 1.0f / se;
    #pragma unroll
    for (int k = 0; k < 7; ++k) out[nidx * 7 + k] = logit[k] * rs;
}

// ---------------------------------------------------------------------------
extern "C" void kernel_launch(void* const* d_in, const int* in_sizes, int n_in,
                              void* d_out, int out_size, void* d_ws, size_t ws_size,
                              hipStream_t stream)
{
    const float* x   = (const float*)d_in[0];
    const int*   M   = (const int*)  d_in[1];
    const float* w   = (const float*)d_in[2];
    const float* w2  = (const float*)d_in[3];
    const float* b2  = (const float*)d_in[4];
    const float* w3  = (const float*)d_in[5];
    const float* b3  = (const float*)d_in[6];
    const float* w4  = (const float*)d_in[7];
    const float* b4  = (const float*)d_in[8];
    const float* w5  = (const float*)d_in[9];
    const float* b5  = (const float*)d_in[10];
    const float* we1 = (const float*)d_in[11];
    const float* be1 = (const float*)d_in[12];
    const float* we2 = (const float*)d_in[13];
    const float* be2 = (const float*)d_in[14];
    const float* we3 = (const float*)d_in[15];
    const float* be3 = (const float*)d_in[16];
    const float* lng = (const float*)d_in[17];
    const float* lnb = (const float*)d_in[18];
    float* out = (float*)d_out;

    // workspace layout (floats): A[256*4096*32] | mean[8192] | inv[8192] | em3[16]
    float* A     = (float*)d_ws;
    float* meanb = A + (size_t)256 * 4096 * 32;
    float* invb  = meanb + 8192;
    float* em3   = invb + 8192;

    emb_kernel<<<1, 128, 0, stream>>>(M, we1, be1, we2, be2, we3, be3, lng, lnb, em3);
    eig_kernel<<<256 * 22, 256, 0, stream>>>(x, A);
    fillemb_kernel<<<4096, 256, 0, stream>>>(A, em3);

    stats_kernel<<<256 * 32, 256, 0, stream>>>(A, meanb, invb);
    resblock_kernel<<<256 * 64, 128, 0, stream>>>(A, meanb, invb, w2, b2, w3, b3);

    stats_kernel<<<256 * 32, 256, 0, stream>>>(A, meanb, invb);
    resblock_kernel<<<256 * 64, 128, 0, stream>>>(A, meanb, invb, w4, b4, w5, b5);

    stats_kernel<<<256 * 32, 256, 0, stream>>>(A, meanb, invb);  // mean == cov3
    final_kernel<<<1, 256, 0, stream>>>(meanb, w, out);
}